// AdaptiveFocalLoss_78520592106026
// MI455X (gfx1250) — compile-verified
//
#include <hip/hip_runtime.h>
#include <math.h>

#define NCLASS   1000
#define CHUNKS   250      // NCLASS / 4 float4 per row
#define WAVESZ   32
#define WAVES    8        // 256 threads / 32
#define GRID     4096

// ---------- wave32 butterfly reductions ----------
__device__ __forceinline__ float wave_max32(float v) {
#pragma unroll
  for (int m = 16; m > 0; m >>= 1) v = fmaxf(v, __shfl_xor(v, m, WAVESZ));
  return v;
}
__device__ __forceinline__ float wave_sum32(float v) {
#pragma unroll
  for (int m = 16; m > 0; m >>= 1) v += __shfl_xor(v, m, WAVESZ);
  return v;
}

// Issue one row (250 float4 chunks) as 8 async global->LDS b128 transfers.
// All 8 share one per-lane base address; the 24-bit INST_OFFSET is applied to
// both the global and the LDS side,
// so only the immediate changes per issue. The 26-chunk tail (chunks 224..249)
// maps exactly to lanes 0..25 of the 8th transfer -> single EXEC predicate.
// Always exactly 8 ASYNCcnt increments per row.
__device__ __forceinline__ void async_prefetch_row(const float* __restrict__ rowp,
                                                   const float4* __restrict__ ldsbuf,
                                                   int lane) {
  const unsigned long long gaddr =
      (unsigned long long)rowp + (unsigned)(lane << 4);
  // low 32 bits of a generic shared pointer == LDS byte offset
  const unsigned lds_off =
      (unsigned)(unsigned long long)ldsbuf + (unsigned)(lane << 4);
#pragma unroll
  for (int k = 0; k < 7; ++k) {
    asm volatile("global_load_async_to_lds_b128 %0, %1, off offset:%c2"
                 :: "v"(lds_off), "v"(gaddr), "i"(k * 512)
                 : "memory");
  }
  if (lane < 26) {
    asm volatile("global_load_async_to_lds_b128 %0, %1, off offset:3584"
                 :: "v"(lds_off), "v"(gaddr)
                 : "memory");
  }
}

__global__ __launch_bounds__(256) void focal_main_kernel(
    const float* __restrict__ x,      // [rows, NCLASS] fp32 logits
    const int*   __restrict__ tgt,    // [rows] class indices
    float*       __restrict__ partial,// [gridDim.x] per-block partial sums
    int rows) {
  __shared__ float4 stage[WAVES][2][CHUNKS];  // 64000 B double-buffered rows
  __shared__ float  wsum[WAVES];

  const int wv      = threadIdx.x >> 5;
  const int lane    = threadIdx.x & 31;
  const int wid     = blockIdx.x * WAVES + wv;
  const int wstride = gridDim.x * WAVES;

  float wacc = 0.0f;
  int buf = 0;

  // ---- prime the async pipeline with this wave's first row ----
  if (wid < rows) {
    async_prefetch_row(x + (long long)wid * NCLASS, &stage[wv][0][0], lane);
  }

  for (int row = wid; row < rows; row += wstride) {
    const int nrow = row + wstride;
    if (nrow < rows) {
      // kick off next row into the other buffer, then wait for current row
      async_prefetch_row(x + (long long)nrow * NCLASS, &stage[wv][buf ^ 1][0], lane);
      asm volatile("s_wait_asynccnt 8" ::: "memory");   // oldest 8 (current row) done
    } else {
      asm volatile("s_wait_asynccnt 0" ::: "memory");   // drain pipeline
    }

    const float4* __restrict__ rb = &stage[wv][buf][0];

    // pass 1: row max (out of LDS)
    float m = -INFINITY;
    for (int c = lane; c < CHUNKS; c += WAVESZ) {
      const float4 v = rb[c];
      m = fmaxf(m, fmaxf(fmaxf(v.x, v.y), fmaxf(v.z, v.w)));
    }
    m = wave_max32(m);

    // pass 2: sum of exp(x - max)
    float s = 0.0f;
    for (int c = lane; c < CHUNKS; c += WAVESZ) {
      const float4 v = rb[c];
      s += __expf(v.x - m) + __expf(v.y - m) + __expf(v.z - m) + __expf(v.w - m);
    }
    s = wave_sum32(s);

    if (lane == 0) {
      const int   t   = tgt[row];
      const float xt  = ((const float*)rb)[t];       // target logit straight from LDS
      const float ce  = (m + __logf(s)) - xt;        // cross-entropy
      const float pt  = __expf(-ce);                 // softmax prob of target
      // gamma table: default 2.0; {1:1.5, 4:3.0, 5:3.5}
      float g = 2.0f;
      if      (t == 1) g = 1.5f;
      else if (t == 4) g = 3.0f;
      else if (t == 5) g = 3.5f;
      const float omp = fmaxf(1.0f - pt, 0.0f);
      wacc += __powf(omp, g) * ce;
    }
    buf ^= 1;
  }

  if (lane == 0) wsum[wv] = wacc;
  __syncthreads();
  if (threadIdx.x == 0) {
    float a = 0.0f;
#pragma unroll
    for (int w = 0; w < WAVES; ++w) a += wsum[w];
    partial[blockIdx.x] = a;
  }
}

// Deterministic final reduction of per-block partials -> mean
__global__ __launch_bounds__(256) void focal_reduce_kernel(
    const float* __restrict__ partial, int n, float inv_n, float* __restrict__ out) {
  __shared__ float sm[256];
  float a = 0.0f;
  for (int i = threadIdx.x; i < n; i += 256) a += partial[i];
  sm[threadIdx.x] = a;
  __syncthreads();
#pragma unroll
  for (int s = 128; s > 0; s >>= 1) {
    if ((int)threadIdx.x < s) sm[threadIdx.x] += sm[threadIdx.x + s];
    __syncthreads();
  }
  if (threadIdx.x == 0) out[0] = sm[0] * inv_n;
}

extern "C" void kernel_launch(void* const* d_in, const int* in_sizes, int n_in,
                              void* d_out, int out_size, void* d_ws, size_t ws_size,
                              hipStream_t stream) {
  (void)n_in; (void)out_size; (void)ws_size;
  const float* x    = (const float*)d_in[0];   // logits [N, 1000] fp32
  const int*   tgt  = (const int*)d_in[1];     // targets [N]
  const int    rows = in_sizes[1];             // N = 262144
  float* partial = (float*)d_ws;               // GRID floats of scratch

  focal_main_kernel<<<dim3(GRID), dim3(256), 0, stream>>>(x, tgt, partial, rows);
  focal_reduce_kernel<<<dim3(1), dim3(256), 0, stream>>>(partial, GRID,
                                                         1.0f / (float)rows,
                                                         (float*)d_out);
}